// Globalizer_7215545057980
// MI455X (gfx1250) — compile-verified
//
#include <hip/hip_runtime.h>
#include <cstdint>

// y[b,n,c,i] = sum_j R[b,n,i,j] * x[b,n,c,j], d=2, C=64, all f32.
// Pure streaming: ~272 MB moved, ~134 MFLOP -> 0.5 FLOP/B, HBM-bound
// (floor ~11.7 us @ 23.3 TB/s). WMMA cannot help; optimize data movement:
//  - gfx1250 async global->LDS b128 loads (ASYNCcnt), double-buffered
//  - non-temporal b128 stores (one-pass output, keep L2 clean)
//  - wave32-aligned chunking so R[p] is a per-wave uniform 16B broadcast load

typedef float v4f __attribute__((ext_vector_type(4)));

#define BLOCK 256
#define ITERS 4

__global__ __launch_bounds__(BLOCK) void rot2d_stream_kernel(
    const float* __restrict__ x, const float* __restrict__ R,
    float* __restrict__ y, long long total /* float4 chunks */)
{
    // Two staging buffers: 2 * 256 threads * 16B = 8 KB LDS per workgroup.
    __shared__ v4f smem[2 * BLOCK];

    const int tid = threadIdx.x;
    const long long stride = (long long)gridDim.x * BLOCK;
    const long long i0 = (long long)blockIdx.x * BLOCK + tid;
    const long long i1 = i0 + stride;
    const long long i2 = i1 + stride;
    const long long i3 = i2 + stride;

    // Group-segment byte addresses of this thread's two LDS slots
    // (low 32 bits of the generic pointer == LDS offset).
    const unsigned lds0 = (unsigned)(uintptr_t)&smem[tid];
    const unsigned lds1 = (unsigned)(uintptr_t)&smem[BLOCK + tid];

    const v4f* __restrict__ xv = (const v4f*)x;
    const v4f* __restrict__ Rv = (const v4f*)R;
    v4f* __restrict__ yv = (v4f*)y;

    // Async copy of one 16B chunk per lane into LDS (tracked by ASYNCcnt).
    auto issue = [&](long long idx, unsigned lds_addr) {
        long long c = idx < total ? idx : total - 1;   // clamp: keep addr valid
        const v4f* gp = xv + c;
        asm volatile("global_load_async_to_lds_b128 %0, %1, off"
                     :: "v"(lds_addr), "v"(gp)
                     : "memory");
    };

    // Consume one staged chunk: ds_load_b128 + 2x2 rotation + NT store.
    auto process = [&](long long idx, int buf) {
        v4f xd = smem[buf * BLOCK + tid];              // ds_load_b128
        long long c = idx < total ? idx : total - 1;
        // 32 consecutive chunks share one point p -> per-wave uniform load.
        v4f r = Rv[c >> 5];                            // (R00,R01,R10,R11)
        v4f o;
        o.x = r.x * xd.x + r.y * xd.y;                 // R00*x0 + R01*x1
        o.y = r.z * xd.x + r.w * xd.y;                 // R10*x0 + R11*x1
        o.z = r.x * xd.z + r.y * xd.w;
        o.w = r.z * xd.z + r.w * xd.w;
        if (idx < total)
            __builtin_nontemporal_store(o, yv + idx);  // th: NT store
    };

    // Double-buffered software pipeline (ASYNCcnt decrements in order,
    // so "<=1" means the older of the two outstanding copies has landed).
    issue(i0, lds0);
    issue(i1, lds1);
    asm volatile("s_wait_asynccnt 1" ::: "memory");
    process(i0, 0);
    issue(i2, lds0);
    asm volatile("s_wait_asynccnt 1" ::: "memory");
    process(i1, 1);
    issue(i3, lds1);
    asm volatile("s_wait_asynccnt 1" ::: "memory");
    process(i2, 0);
    asm volatile("s_wait_asynccnt 0" ::: "memory");
    process(i3, 1);
}

extern "C" void kernel_launch(void* const* d_in, const int* in_sizes, int n_in,
                              void* d_out, int out_size, void* d_ws, size_t ws_size,
                              hipStream_t stream) {
    (void)n_in; (void)out_size; (void)d_ws; (void)ws_size;
    const float* x = (const float*)d_in[0];   // [64,4096,128] f32
    const float* R = (const float*)d_in[1];   // [64,4096,2,2] f32
    float* y = (float*)d_out;                 // [64,4096,128] f32

    long long total = (long long)in_sizes[0] / 4;     // float4 chunks (2^23)
    const long long per_block = (long long)BLOCK * ITERS;
    int grid = (int)((total + per_block - 1) / per_block);   // 8192 expected
    if (grid < 1) grid = 1;

    rot2d_stream_kernel<<<grid, BLOCK, 0, stream>>>(x, R, y, total);
}